// MultiHeadAttentionLayerSANSparse_6081673691509
// MI455X (gfx1250) — compile-verified
//
#include <hip/hip_runtime.h>
#include <hip/hip_bf16.h>

// ---------------------------------------------------------------------------
// Sparse multi-head graph attention (SAN) for MI455X / gfx1250.
// N=50000 nodes, E=800000 edges, IN=64, H=8, D=8 (HD=64).
// Memory-bound: edge_attr stream (205 MB, loaded NT) dominates; Q/K/V tables,
// score buffer and output (~90 MB) stay L2-resident (192 MB), so gathers and
// scatter-atomics are L2 traffic, not HBM.
// GEMMs via v_wmma_f32_16x16x4_f32 (exact fp32 path, K=64 -> 16 chained WMMAs):
// at ~6.5 GFLOP total the matrix pipe stays below the 9 us HBM floor even at
// 1/8 the f16 WMMA rate, so full-precision math is free.
// ---------------------------------------------------------------------------

typedef __attribute__((ext_vector_type(16))) _Float16 v16h;
typedef __attribute__((ext_vector_type(8)))  float    v8f;
typedef __attribute__((ext_vector_type(4)))  float    v4f;
typedef __attribute__((ext_vector_type(2)))  float    v2f;

#define IN_DIM 64
#define HD     64
#define CLAMP_V 5.0f
#define MAX_BIAS 16.0f   // score in [-5,5] -> score+16 > 0: float bits monotone as int

// V_WMMA_F32_16X16X4_F32: D[16,16] = A[16,4] x B[4,16] + C, all fp32.
// A layout (ISA 7.12.2, 32-bit A 16x4): lane l<16 -> {A[l,0],A[l,1]},
// lane l+16 -> {A[l,2],A[l,3]}.  B mirrors it column-wise: lane n<16 ->
// {B[0,n],B[1,n]}, lane n+16 -> {B[2,n],B[3,n]}.  C/D: v8f, VGPR r -> row
// r + 8*hi, col = lane&15.
__device__ __forceinline__ v8f wmma_f32(v2f a, v2f b, v8f c) {
  // (neg_a, A, neg_b, B, c_mod, C, reuse_a, reuse_b)
  return __builtin_amdgcn_wmma_f32_16x16x4_f32(false, a, false, b, (short)0, c,
                                               false, false);
}

// Per-lane A fragments for a full K=64 row: 16 chunks of float2 at
// row[kc*4 + hi*2].  Non-temporal: x/edge_attr are streamed exactly once and
// must not evict the L2-resident gather tables.
__device__ __forceinline__ void load_a_frags_nt(const float* __restrict__ row,
                                                int hi, v2f* __restrict__ a) {
#pragma unroll
  for (int kc = 0; kc < 16; ++kc)
    a[kc] = __builtin_nontemporal_load((const v2f*)(row + kc * 4 + hi * 2));
}

// ---------------------------------------------------------------------------
// K0: initialize output, denominators, segment-max bits.
// ---------------------------------------------------------------------------
__global__ void san_init_kernel(float* __restrict__ out,
                                float* __restrict__ denom,
                                int* __restrict__ segmax, int n_nodes) {
  int tid = blockIdx.x * blockDim.x + threadIdx.x;
  if (tid < n_nodes * HD) out[tid] = 0.0f;
  if (tid < n_nodes * 8) {
    denom[tid]  = 0.0f;
    segmax[tid] = 0;  // bits of +0.0f, below any (score+16) in [11,21]
  }
}

// ---------------------------------------------------------------------------
// K1: node projections Q,K,V = x @ W^T + b.  One wave per 16-node tile.
// ---------------------------------------------------------------------------
__global__ void __launch_bounds__(256)
san_node_proj_kernel(const float* __restrict__ x,
                     const float* __restrict__ Qw, const float* __restrict__ Qb,
                     const float* __restrict__ Kw, const float* __restrict__ Kb,
                     const float* __restrict__ Vw, const float* __restrict__ Vb,
                     float* __restrict__ Qh, float* __restrict__ Kh,
                     float* __restrict__ Vh, int n_nodes) {
  const int wave = (blockIdx.x * blockDim.x + threadIdx.x) >> 5;
  const int lane = threadIdx.x & 31;
  if (wave * 16 >= n_nodes) return;                 // whole-wave uniform exit
  const int m  = lane & 15;
  const int hi = lane >> 4;
  const bool full = (wave * 16 + 16 <= n_nodes);    // wave-uniform
  int row = wave * 16 + m;
  if (row >= n_nodes) row = n_nodes - 1;
  const float* xrow = x + (size_t)row * IN_DIM;

  v2f a[16];
  load_a_frags_nt(xrow, hi, a);

  const float* Ws[3] = {Qw, Kw, Vw};
  const float* Bs[3] = {Qb, Kb, Vb};
  float*       Os[3] = {Qh, Kh, Vh};

#pragma unroll
  for (int w = 0; w < 3; ++w) {
#pragma unroll
    for (int oc = 0; oc < 4; ++oc) {
      const int col = oc * 16 + m;                  // this lane's output column
      const float* wrow = Ws[w] + (size_t)col * IN_DIM;
      v8f c = {};
#pragma unroll
      for (int kc = 0; kc < 16; ++kc) {
        const v2f b = *(const v2f*)(wrow + kc * 4 + hi * 2);
        c = wmma_f32(a[kc], b, c);
      }
      const float bias = Bs[w][col];
      // D layout: VGPR r -> row (r + 8*hi), col = this lane's col
      float* obase = Os[w] + (size_t)(wave * 16 + 8 * hi) * HD + col;
      if (full) {
#pragma unroll
        for (int r = 0; r < 8; ++r) obase[r * HD] = c[r] + bias;
      } else {
#pragma unroll
        for (int r = 0; r < 8; ++r)
          if (wave * 16 + 8 * hi + r < n_nodes) obase[r * HD] = c[r] + bias;
      }
    }
  }
}

// ---------------------------------------------------------------------------
// K2: edge projection (WMMA) fused with per-edge attention score + segment max.
// One wave per 16-edge tile.
// ---------------------------------------------------------------------------
__global__ void __launch_bounds__(256)
san_edge_score_kernel(const float* __restrict__ edge_attr,
                      const int* __restrict__ src_idx,
                      const int* __restrict__ dst_idx,
                      const float* __restrict__ Ew, const float* __restrict__ Eb,
                      const float* __restrict__ Kh, const float* __restrict__ Qh,
                      float* __restrict__ score, int* __restrict__ segmax,
                      int n_edges) {
  const int wave = (blockIdx.x * blockDim.x + threadIdx.x) >> 5;
  const int lane = threadIdx.x & 31;
  if (wave * 16 >= n_edges) return;                 // whole-wave uniform exit
  const int m  = lane & 15;
  const int hi = lane >> 4;
  const bool full = (wave * 16 + 16 <= n_edges);    // wave-uniform
  int erow = wave * 16 + m;
  if (erow >= n_edges) erow = n_edges - 1;
  const float* arow = edge_attr + (size_t)erow * IN_DIM;

  v2f a[16];
  load_a_frags_nt(arow, hi, a);

  const int my_src = src_idx[erow];                 // lane m holds edge m's indices
  const int my_dst = dst_idx[erow];
  const float inv_sqrt_d = 0.35355339059327373f;    // 1/sqrt(8)

  // Hoisted: this lane's view of the tile's src/dst indices for rows
  // em = r + 8*hi (oc-invariant, saves 48 lane-permutes).
  int s_em[8], d_em[8];
#pragma unroll
  for (int r = 0; r < 8; ++r) {
    s_em[r] = __shfl(my_src, r + 8 * hi);
    d_em[r] = __shfl(my_dst, r + 8 * hi);
  }

#pragma unroll
  for (int oc = 0; oc < 4; ++oc) {
    const int out = oc * 16 + m;                    // this lane's HD column
    const float* wrow = Ew + (size_t)out * IN_DIM;
    v8f c = {};
#pragma unroll
    for (int kc = 0; kc < 16; ++kc) {
      const v2f b = *(const v2f*)(wrow + kc * 4 + hi * 2);
      c = wmma_f32(a[kc], b, c);
    }
    const float bias = Eb[out];
    const int h = out >> 3;                         // head of this lane's column

#pragma unroll
    for (int r = 0; r < 8; ++r) {
      const int em = r + 8 * hi;                    // edge row held in c[r]
      const float ew = c[r] + bias;                 // E_w[edge, out]
      float val = ew * Kh[(size_t)s_em[r] * HD + out] *
                  Qh[(size_t)d_em[r] * HD + out] * inv_sqrt_d;
      // reduce the 8 lanes covering d=0..7 of this head
      val += __shfl_xor(val, 1);
      val += __shfl_xor(val, 2);
      val += __shfl_xor(val, 4);
      if ((m & 7) == 0 && (full || wave * 16 + em < n_edges)) {
        const float sc = fminf(fmaxf(val, -CLAMP_V), CLAMP_V);
        const int edge = wave * 16 + em;
        score[(size_t)edge * 8 + h] = sc;
        // positive-float bits are monotone as ints -> int atomicMax works
        atomicMax(&segmax[(size_t)d_em[r] * 8 + h],
                  __float_as_int(sc + MAX_BIAS));
      }
    }
  }
}

// ---------------------------------------------------------------------------
// K3: ex = exp(score - segmax[dst]); denom[dst] += ex  (segment softmax pt.1)
// One thread per edge: 1 dst load, 8 contiguous score elems, 8 atomics.
// ---------------------------------------------------------------------------
__global__ void san_softmax_kernel(const int* __restrict__ dst_idx,
                                   const int* __restrict__ segmax,
                                   float* __restrict__ score,  // in: score, out: ex
                                   float* __restrict__ denom, int n_edges) {
  int e = blockIdx.x * blockDim.x + threadIdx.x;
  if (e >= n_edges) return;
  const int d = dst_idx[e];
  const int* mxp  = segmax + (size_t)d * 8;
  float*     sp   = score + (size_t)e * 8;
  float*     dp   = denom + (size_t)d * 8;
#pragma unroll
  for (int h = 0; h < 8; ++h) {
    const float mx = __int_as_float(mxp[h]) - MAX_BIAS;
    const float ex = __expf(sp[h] - mx);
    sp[h] = ex;
    atomicAdd(dp + h, ex);
  }
}

// ---------------------------------------------------------------------------
// K4: out[dst] += V[src] * ex/denom[dst]   (weighted scatter-add)
// 16 threads per edge, float4 V load, 4 atomics each -> coalesced bursts into
// the L2-resident output.
// ---------------------------------------------------------------------------
__global__ void san_aggregate_kernel(const int* __restrict__ src_idx,
                                     const int* __restrict__ dst_idx,
                                     const float* __restrict__ Vh,
                                     const float* __restrict__ ex,
                                     const float* __restrict__ denom,
                                     float* __restrict__ out, int n_edges) {
  int tid = blockIdx.x * blockDim.x + threadIdx.x;
  if (tid >= n_edges * 16) return;
  const int e = tid >> 4;
  const int o = (tid & 15) * 4;     // 4 consecutive outputs
  const int h = o >> 3;
  const int s = src_idx[e];
  const int d = dst_idx[e];
  const float attn =
      ex[(size_t)e * 8 + h] / (denom[(size_t)d * 8 + h] + 1e-16f);
  const v4f v = *(const v4f*)(Vh + (size_t)s * HD + o);
  float* po = out + (size_t)d * HD + o;
  atomicAdd(po + 0, v[0] * attn);
  atomicAdd(po + 1, v[1] * attn);
  atomicAdd(po + 2, v[2] * attn);
  atomicAdd(po + 3, v[3] * attn);
}

// ---------------------------------------------------------------------------
extern "C" void kernel_launch(void* const* d_in, const int* in_sizes, int n_in,
                              void* d_out, int out_size, void* d_ws,
                              size_t ws_size, hipStream_t stream) {
  const float* x         = (const float*)d_in[0];
  const float* edge_attr = (const float*)d_in[1];
  const int*   edge_idx  = (const int*)d_in[2];   // [2,E]: row0=src, row1=dst
  const float* Qw = (const float*)d_in[3];
  const float* Qb = (const float*)d_in[4];
  const float* Kw = (const float*)d_in[5];
  const float* Kb = (const float*)d_in[6];
  const float* Ew = (const float*)d_in[7];
  const float* Eb = (const float*)d_in[8];
  const float* Vw = (const float*)d_in[9];
  const float* Vb = (const float*)d_in[10];
  float* out = (float*)d_out;

  const int N = in_sizes[0] / IN_DIM;
  const int E = in_sizes[1] / IN_DIM;
  const int* src_idx = edge_idx;
  const int* dst_idx = edge_idx + E;

  // workspace layout (~67 MB total)
  float* Qh    = (float*)d_ws;                 // N*64
  float* Kh    = Qh + (size_t)N * HD;          // N*64
  float* Vh    = Kh + (size_t)N * HD;          // N*64
  float* score = Vh + (size_t)N * HD;          // E*8 (score, then ex)
  float* denom = score + (size_t)E * 8;        // N*8
  int*   segmax = (int*)(denom + (size_t)N * 8);  // N*8

  const int TB = 256;

  {  // K0: init
    int t = N * HD;
    san_init_kernel<<<(t + TB - 1) / TB, TB, 0, stream>>>(out, denom, segmax, N);
  }
  {  // K1: node projections
    int waves = (N + 15) / 16;
    int th = waves * 32;
    san_node_proj_kernel<<<(th + TB - 1) / TB, TB, 0, stream>>>(
        x, Qw, Qb, Kw, Kb, Vw, Vb, Qh, Kh, Vh, N);
  }
  {  // K2: edge projection + scores + segment max
    int waves = (E + 15) / 16;
    int th = waves * 32;
    san_edge_score_kernel<<<(th + TB - 1) / TB, TB, 0, stream>>>(
        edge_attr, src_idx, dst_idx, Ew, Eb, Kh, Qh, score, segmax, E);
  }
  {  // K3: exp + segment sum
    san_softmax_kernel<<<(E + TB - 1) / TB, TB, 0, stream>>>(dst_idx, segmax,
                                                             score, denom, E);
  }
  {  // K4: weighted scatter-add
    int t = E * 16;
    san_aggregate_kernel<<<(t + TB - 1) / TB, TB, 0, stream>>>(
        src_idx, dst_idx, Vh, score, denom, out, E);
  }
}